// CandidateRepresentationLayer_11158325035066
// MI455X (gfx1250) — compile-verified
//
#include <hip/hip_runtime.h>

typedef __attribute__((ext_vector_type(2))) float v2f;
typedef __attribute__((ext_vector_type(8))) float v8f;

#define BATCH 32
#define SLEN  512
#define DIM   1024
#define ANCH  4
#define NCAND 65536

// One wave32 per candidate. Mean-pool of <=8 rows expressed as
// D = A(16x4) x B(4x16) + C with replicated weights A[m,k] = w_k,
// chained twice (k=0..3, k=4..7). Every row of D equals the span mean
// for one 16-column chunk; we do two chunks per iteration so the low
// half-wave stores chunk c and the high half-wave stores chunk c+1
// (full 128B coalesced non-temporal store).
__global__ __launch_bounds__(32) void pool_wmma_kernel(
    const float* __restrict__ word_repr,
    const int*   __restrict__ cand,
    const int*   __restrict__ anchor_loc,
    float*       __restrict__ out_repr)
{
    const int n    = blockIdx.x;
    const int lane = threadIdx.x;            // 0..31

    // Wave-uniform candidate indices (compiler emits scalar loads).
    const int b = cand[3 * n + 0];
    const int w = cand[3 * n + 1];
    const int a = cand[3 * n + 2];
    const int locbase = ((b * SLEN + w) * ANCH + a) * 2;
    const int sid = anchor_loc[locbase + 0];
    const int eid = anchor_loc[locbase + 1];
    const int len = eid - sid;               // 0..8
    const float inv = (len > 0) ? (1.0f / (float)len) : 0.0f;

    // Half-wave K assignment: lanes 0-15 -> k={0,1}, lanes 16-31 -> k={2,3}
    // (and +4 for the second WMMA).
    const int rbase = (lane >> 4) * 2;
    const int col   = (lane & 15);

    // Replicated-weight A operands (w_k = inv if k < len else 0).
    v2f a0, a1;
    a0.x = (rbase + 0 < len) ? inv : 0.0f;
    a0.y = (rbase + 1 < len) ? inv : 0.0f;
    a1.x = (rbase + 4 < len) ? inv : 0.0f;
    a1.y = (rbase + 5 < len) ? inv : 0.0f;

    // Row indices, clamped to sid for weight-0 rows (address always valid;
    // sid < SLEN always, and sid+k < eid <= SLEN when k < len).
    const int r0 = sid + ((rbase + 0 < len) ? (rbase + 0) : 0);
    const int r1 = sid + ((rbase + 1 < len) ? (rbase + 1) : 0);
    const int r4 = sid + ((rbase + 4 < len) ? (rbase + 4) : 0);
    const int r5 = sid + ((rbase + 5 < len) ? (rbase + 5) : 0);

    const size_t base = (size_t)b * SLEN * DIM;
    const float* __restrict__ row0 = word_repr + base + (size_t)r0 * DIM;
    const float* __restrict__ row1 = word_repr + base + (size_t)r1 * DIM;
    const float* __restrict__ row4 = word_repr + base + (size_t)r4 * DIM;
    const float* __restrict__ row5 = word_repr + base + (size_t)r5 * DIM;

    float* __restrict__ orow = out_repr + (size_t)n * DIM;

    #pragma unroll 4
    for (int c = 0; c < DIM / 16; c += 2) {
        const int j0 = c * 16 + col;
        const int j1 = j0 + 16;

        v2f b00, b01, b10, b11;
        b00.x = row0[j0]; b00.y = row1[j0];   // k = rbase, rbase+1
        b01.x = row4[j0]; b01.y = row5[j0];   // k = rbase+4, rbase+5
        b10.x = row0[j1]; b10.y = row1[j1];
        b11.x = row4[j1]; b11.y = row5[j1];

        v8f acc0 = {};
        v8f acc1 = {};
        acc0 = __builtin_amdgcn_wmma_f32_16x16x4_f32(false, a0, false, b00,
                                                     (short)0, acc0, false, false);
        acc0 = __builtin_amdgcn_wmma_f32_16x16x4_f32(false, a1, false, b01,
                                                     (short)0, acc0, false, false);
        acc1 = __builtin_amdgcn_wmma_f32_16x16x4_f32(false, a0, false, b10,
                                                     (short)0, acc1, false, false);
        acc1 = __builtin_amdgcn_wmma_f32_16x16x4_f32(false, a1, false, b11,
                                                     (short)0, acc1, false, false);

        // All 16 rows of D are identical; lane L holds column (L & 15) of its
        // chunk in acc[0]. Low half stores chunk c, high half chunk c+1:
        // global column = c*16 + lane for every lane -> full-wave 128B store.
        const float v = (lane < 16) ? acc0[0] : acc1[0];
        __builtin_nontemporal_store(v, &orow[c * 16 + lane]);
    }
}

__global__ void zero_counts_kernel(float* __restrict__ counts)
{
    const int i = threadIdx.x;
    if (i < BATCH) counts[i] = 0.0f;
}

__global__ __launch_bounds__(256) void meta_kernel(
    const int* __restrict__ cand,
    const int* __restrict__ anchor_loc,
    const int* __restrict__ anchor_cls,
    float* __restrict__ out_label,
    float* __restrict__ out_counts,
    float* __restrict__ out_valid,
    float* __restrict__ out_loc)
{
    __shared__ float lcnt[BATCH];
    if (threadIdx.x < BATCH) lcnt[threadIdx.x] = 0.0f;
    __syncthreads();

    const int n = blockIdx.x * blockDim.x + threadIdx.x;
    if (n < NCAND) {
        const int b = cand[3 * n + 0];
        const int w = cand[3 * n + 1];
        const int a = cand[3 * n + 2];
        const int locbase = ((b * SLEN + w) * ANCH + a) * 2;
        const int sid = anchor_loc[locbase + 0];
        const int eid = anchor_loc[locbase + 1];
        const bool valid = eid > sid;

        out_label[n]       = valid ? (float)anchor_cls[(b * SLEN + w) * ANCH + a] : -1.0f;
        out_valid[n]       = valid ? 1.0f : 0.0f;
        out_loc[2 * n + 0] = valid ? (float)sid : 0.0f;
        out_loc[2 * n + 1] = valid ? (float)eid : 0.0f;
        if (valid) atomicAdd(&lcnt[b], 1.0f);
    }
    __syncthreads();
    if (threadIdx.x < BATCH) {
        const float c = lcnt[threadIdx.x];
        if (c != 0.0f) atomicAdd(&out_counts[threadIdx.x], c);
    }
}

extern "C" void kernel_launch(void* const* d_in, const int* in_sizes, int n_in,
                              void* d_out, int out_size, void* d_ws, size_t ws_size,
                              hipStream_t stream) {
    (void)in_sizes; (void)n_in; (void)out_size; (void)d_ws; (void)ws_size;

    const float* word_repr  = (const float*)d_in[0];
    const int*   cand       = (const int*)d_in[1];
    const int*   anchor_loc = (const int*)d_in[2];
    const int*   anchor_cls = (const int*)d_in[3];

    float* out = (float*)d_out;
    // Concatenated tuple layout (all float32):
    // [repr N*D][label N][counts B][valid N][loc N*2]
    const size_t o_label  = (size_t)NCAND * DIM;
    const size_t o_counts = o_label + NCAND;
    const size_t o_valid  = o_counts + BATCH;
    const size_t o_loc    = o_valid + NCAND;

    float* out_repr   = out;
    float* out_label  = out + o_label;
    float* out_counts = out + o_counts;
    float* out_valid  = out + o_valid;
    float* out_loc    = out + o_loc;

    zero_counts_kernel<<<1, 32, 0, stream>>>(out_counts);
    meta_kernel<<<(NCAND + 255) / 256, 256, 0, stream>>>(
        cand, anchor_loc, anchor_cls, out_label, out_counts, out_valid, out_loc);
    pool_wmma_kernel<<<NCAND, 32, 0, stream>>>(
        word_repr, cand, anchor_loc, out_repr);
}